// ClusterGCNLayer_3925600109367
// MI455X (gfx1250) — compile-verified
//
#include <hip/hip_runtime.h>

typedef __attribute__((ext_vector_type(2))) float v2f;
typedef __attribute__((ext_vector_type(8))) float v8f;

#define DIM 128
#define NCLUST 64

// ---- init: zero agg, degree counters, cluster counters ----
__global__ void k_init(float* __restrict__ agg, unsigned* __restrict__ degc,
                       unsigned* __restrict__ ccnt, long long total) {
  long long gid = (long long)blockIdx.x * blockDim.x + threadIdx.x;
  if (gid < total) agg[gid] = 0.0f;
  if (gid < total / DIM) degc[gid] = 0u;      // total/DIM == n_nodes
  if (gid < NCLUST) ccnt[gid] = 0u;
}

// ---- GEMM: XW = X @ W  via V_WMMA_F32_16X16X4_F32 ----
// One block = 8 waves; block owns 16 rows of X (staged in LDS), wave w owns
// columns [16w, 16w+16) of W. 32 WMMA steps cover K=128.
__global__ void k_gemm(const float* __restrict__ X, const float* __restrict__ W,
                       float* __restrict__ XW, int n_nodes) {
  __shared__ float sX[16 * DIM];              // 8 KB
  const int tid  = threadIdx.x;               // 0..255
  const int wave = tid >> 5;                  // 0..7
  const int lane = tid & 31;
  const int half = lane >> 4;                 // 0 or 1
  const int lrow = lane & 15;
  const int m0   = blockIdx.x << 4;

  // coalesced stage of the 16x128 X slab
  for (int i = tid; i < 16 * DIM; i += 256) {
    int r = m0 + (i >> 7);
    if (r >= n_nodes) r = n_nodes - 1;        // clamp (keeps EXEC uniform)
    sX[i] = X[(long long)r * DIM + (i & 127)];
  }
  __syncthreads();

  const int n0  = wave << 4;
  const int col = n0 + lrow;
  v8f acc = {0.f, 0.f, 0.f, 0.f, 0.f, 0.f, 0.f, 0.f};

#pragma unroll
  for (int k0 = 0; k0 < DIM; k0 += 4) {
    const int ka = k0 + (half << 1);          // half 0 -> K{0,1}, half 1 -> K{2,3}
    v2f a, b;
    a.x = sX[lrow * DIM + ka];                // A: row m0+lrow, K=ka
    a.y = sX[lrow * DIM + ka + 1];
    b.x = W[ka * DIM + col];                  // B: row ka, col n0+lrow
    b.y = W[(ka + 1) * DIM + col];
    acc = __builtin_amdgcn_wmma_f32_16x16x4_f32(
        /*neg_a=*/false, a, /*neg_b=*/false, b,
        /*c_mod=*/(short)0, acc, /*reuse_a=*/false, /*reuse_b=*/false);
  }

#pragma unroll
  for (int i = 0; i < 8; ++i) {               // VGPR i: M = i (lanes 0-15) / 8+i (16-31)
    int row = m0 + (half << 3) + i;
    if (row < n_nodes) XW[(long long)row * DIM + col] = acc[i];
  }
}

// ---- degree + per-cluster intra-edge count (LDS-reduced histogram) ----
__global__ void k_degree(const int* __restrict__ ei, const int* __restrict__ ca,
                         unsigned* __restrict__ degc, unsigned* __restrict__ ccnt,
                         int n_edges) {
  __shared__ unsigned h[NCLUST];
  if (threadIdx.x < NCLUST) h[threadIdx.x] = 0u;
  __syncthreads();
  int e = blockIdx.x * blockDim.x + threadIdx.x;
  if (e < n_edges) {
    int s = ei[e];
    int d = ei[n_edges + e];
    int cs = ca[s];
    if (cs == ca[d]) {
      atomicAdd(&degc[d], 1u);
      atomicAdd(&h[cs], 1u);
    }
  }
  __syncthreads();
  if (threadIdx.x < NCLUST) {
    unsigned v = h[threadIdx.x];
    if (v) atomicAdd(&ccnt[threadIdx.x], v);
  }
}

// ---- dinv[n] = rsqrt(deg[n]) with deg = count + 1 (self loop) ----
__global__ void k_dinv(const unsigned* __restrict__ degc, float* __restrict__ dinv,
                       int n_nodes) {
  int n = blockIdx.x * blockDim.x + threadIdx.x;
  if (n < n_nodes) dinv[n] = rsqrtf((float)(degc[n] + 1u));
}

// ---- edge scatter: agg[dst] += dinv[s]*dinv[d] * xw[src]  (128 lanes/edge) ----
__global__ void k_scatter(const int* __restrict__ ei, const int* __restrict__ ca,
                          const float* __restrict__ dinv, const float* __restrict__ xw,
                          float* __restrict__ agg, int n_edges) {
  int e = blockIdx.x * 2 + (threadIdx.x >> 7);
  int c = threadIdx.x & 127;
  if (e >= n_edges) return;
  int s = ei[e];
  int d = ei[n_edges + e];
  if (ca[s] != ca[d]) return;
  float norm = dinv[s] * dinv[d];
  atomicAdd(&agg[(long long)d * DIM + c], norm * xw[(long long)s * DIM + c]);
}

// ---- finalize: out = has_edges ? agg + xw/deg + b : X ----
__global__ void k_final(const float* __restrict__ X, const float* __restrict__ b,
                        const int* __restrict__ ca, const unsigned* __restrict__ degc,
                        const unsigned* __restrict__ ccnt, const float* __restrict__ xw,
                        const float* __restrict__ agg, float* __restrict__ out,
                        int n_nodes) {
  long long gid = (long long)blockIdx.x * blockDim.x + threadIdx.x;
  long long total = (long long)n_nodes * DIM;
  if (gid >= total) return;
  int n = (int)(gid >> 7);
  int c = (int)(gid & 127);
  bool has = ccnt[ca[n]] > 0u;
  float invdeg = 1.0f / (float)(degc[n] + 1u);
  float v = agg[gid] + xw[gid] * invdeg + b[c];
  out[gid] = has ? v : X[gid];
}

extern "C" void kernel_launch(void* const* d_in, const int* in_sizes, int n_in,
                              void* d_out, int out_size, void* d_ws, size_t ws_size,
                              hipStream_t stream) {
  const float* X  = (const float*)d_in[0];
  const float* W  = (const float*)d_in[1];
  const float* b  = (const float*)d_in[2];
  // d_in[3] = edge_attr: unused by the reference forward
  const int*   ca = (const int*)d_in[4];
  const int*   ei = (const int*)d_in[5];

  const int n_nodes = in_sizes[0] / DIM;
  const int n_edges = in_sizes[5] / 2;
  const long long total = (long long)n_nodes * DIM;

  // workspace carve-up
  char* ws = (char*)d_ws;
  size_t off = 0;
  float*    xw   = (float*)(ws + off);    off += (size_t)total * sizeof(float);
  float*    agg  = (float*)(ws + off);    off += (size_t)total * sizeof(float);
  unsigned* degc = (unsigned*)(ws + off); off += (((size_t)n_nodes * 4) + 255) & ~(size_t)255;
  unsigned* ccnt = (unsigned*)(ws + off); off += 256;
  float*    dinv = (float*)(ws + off);    off += (size_t)n_nodes * sizeof(float);
  (void)ws_size; (void)n_in; (void)out_size;

  const int blk = 256;
  const int g_elem = (int)((total + blk - 1) / blk);

  hipLaunchKernelGGL(k_init, dim3(g_elem), dim3(blk), 0, stream, agg, degc, ccnt, total);

  const int g_gemm = (n_nodes + 15) / 16;
  hipLaunchKernelGGL(k_gemm, dim3(g_gemm), dim3(256), 0, stream, X, W, xw, n_nodes);

  const int g_deg = (n_edges + blk - 1) / blk;
  hipLaunchKernelGGL(k_degree, dim3(g_deg), dim3(blk), 0, stream, ei, ca, degc, ccnt, n_edges);

  const int g_dinv = (n_nodes + blk - 1) / blk;
  hipLaunchKernelGGL(k_dinv, dim3(g_dinv), dim3(blk), 0, stream, degc, dinv, n_nodes);

  const int g_sc = (n_edges + 1) / 2;
  hipLaunchKernelGGL(k_scatter, dim3(g_sc), dim3(256), 0, stream, ei, ca, dinv, xw, agg, n_edges);

  hipLaunchKernelGGL(k_final, dim3(g_elem), dim3(blk), 0, stream,
                     X, b, ca, degc, ccnt, xw, agg, (float*)d_out, n_nodes);
}